// PyG_SGC_Paper_15874199126235
// MI455X (gfx1250) — compile-verified
//
#include <hip/hip_runtime.h>
#include <hip/hip_bf16.h>

typedef __attribute__((ext_vector_type(2))) float v2f;
typedef __attribute__((ext_vector_type(8))) float v8f;

#define N_NODES 50000
#define N_EDGES 600000
#define D 128
#define LDSTRIDE 132  // 128 + 4 pad: bank stride 4 -> conflict-free 16-lane column reads

// ---------- degree / dinv ----------
__global__ void k_deg_init(float* __restrict__ deg, int n) {
    int i = blockIdx.x * blockDim.x + threadIdx.x;
    if (i < n) deg[i] = 1.0f;  // self loop contributes 1
}

__global__ void k_deg_count(const int* __restrict__ dst, float* __restrict__ deg, int ne) {
    int e = blockIdx.x * blockDim.x + threadIdx.x;
    if (e < ne) atomicAdd(&deg[dst[e]], 1.0f);
}

__global__ void k_dinv(float* __restrict__ deg, int n) {
    int i = blockIdx.x * blockDim.x + threadIdx.x;
    if (i < n) {
        float d = deg[i];
        deg[i] = (d > 0.0f) ? rsqrtf(d) : 0.0f;  // in-place: deg -> dinv
    }
}

// ---------- propagation: self-loop term initializes Hout ----------
// Hout[i][:] = dinv[i]^2 * Hin[i][:]   (one wave-lane handles a float4)
__global__ void k_self_init(const float* __restrict__ Hin, const float* __restrict__ dinv,
                            float* __restrict__ Hout, int n) {
    int idx = blockIdx.x * blockDim.x + threadIdx.x;
    int node = idx >> 5;
    int q = idx & 31;
    if (node >= n) return;
    float s = dinv[node];
    s *= s;
    float4 v = ((const float4*)(Hin + (size_t)node * D))[q];
    float4 o;
    o.x = s * v.x; o.y = s * v.y; o.z = s * v.z; o.w = s * v.w;
    ((float4*)(Hout + (size_t)node * D))[q] = o;
}

// ---------- propagation: edge scatter (1 edge per wave32, float4 per lane) ----------
__global__ void k_edge_scatter(const int* __restrict__ src, const int* __restrict__ dst,
                               const float* __restrict__ dinv,
                               const float* __restrict__ Hin, float* __restrict__ Hout, int ne) {
    int gid = blockIdx.x * blockDim.x + threadIdx.x;
    int e = gid >> 5;
    int lane = threadIdx.x & 31;
    if (e >= ne) return;
    int s = src[e];
    int d = dst[e];
    float nrm = dinv[s] * dinv[d];
    float4 v = ((const float4*)(Hin + (size_t)s * D))[lane];  // 32 lanes x 16B = full 512B row
    float* o = Hout + (size_t)d * D + lane * 4;
    atomicAdd(o + 0, nrm * v.x);
    atomicAdd(o + 1, nrm * v.y);
    atomicAdd(o + 2, nrm * v.z);
    atomicAdd(o + 3, nrm * v.w);
}

// ---------- final GEMM: Out = H @ W^T + b  via V_WMMA_F32_16X16X4_F32 ----------
// Block: 256 threads = 8 waves, computes a 64(row) x 128(col) output tile.
// wave -> row-tile rt = wave&3 (16 rows), col-half ch = wave>>2 (4 x 16-col tiles).
__global__ void k_gemm_wmma(const float* __restrict__ H, const float* __restrict__ W,
                            const float* __restrict__ bias, float* __restrict__ out, int n) {
    __shared__ float sH[64 * LDSTRIDE];  // 33.8 KB

    int tid = threadIdx.x;
    int mbase = blockIdx.x * 64;

    // Cooperative, coalesced load of 64x128 H tile (zero-pad past n)
    for (int i = tid; i < 64 * 32; i += 256) {
        int r = i >> 5;   // row within tile
        int c = i & 31;   // float4 column
        int grow = mbase + r;
        float4 v;
        if (grow < n) v = ((const float4*)(H + (size_t)grow * D))[c];
        else { v.x = 0.f; v.y = 0.f; v.z = 0.f; v.w = 0.f; }
        float* p = &sH[r * LDSTRIDE + c * 4];
        p[0] = v.x; p[1] = v.y; p[2] = v.z; p[3] = v.w;
    }
    __syncthreads();

    int wave = tid >> 5;
    int lane = tid & 31;
    int half = lane >> 4;   // 0: K=k0,k0+1   1: K=k0+2,k0+3
    int l16  = lane & 15;   // M (for A) / N (for B,D) within tile
    int rt = wave & 3;      // row tile (16 rows each)
    int ch = wave >> 2;     // column half (64 cols each)

    v8f acc0 = {}, acc1 = {}, acc2 = {}, acc3 = {};

    // B fragments: B[k][n] = W[n][k]; this lane's n rows of W for the 4 col-tiles
    const float* wb = W + (size_t)(ch * 64 + l16) * D;
    const float* ar = &sH[(rt * 16 + l16) * LDSTRIDE];

    for (int k0 = 0; k0 < D; k0 += 4) {
        int kk = k0 + 2 * half;
        v2f a;
        a.x = ar[kk];
        a.y = ar[kk + 1];
        v2f b0, b1, b2, b3;
        b0.x = wb[0 * 16 * D + kk]; b0.y = wb[0 * 16 * D + kk + 1];
        b1.x = wb[1 * 16 * D + kk]; b1.y = wb[1 * 16 * D + kk + 1];
        b2.x = wb[2 * 16 * D + kk]; b2.y = wb[2 * 16 * D + kk + 1];
        b3.x = wb[3 * 16 * D + kk]; b3.y = wb[3 * 16 * D + kk + 1];
        acc0 = __builtin_amdgcn_wmma_f32_16x16x4_f32(false, a, false, b0, (short)0, acc0, false, false);
        acc1 = __builtin_amdgcn_wmma_f32_16x16x4_f32(false, a, false, b1, (short)0, acc1, false, false);
        acc2 = __builtin_amdgcn_wmma_f32_16x16x4_f32(false, a, false, b2, (short)0, acc2, false, false);
        acc3 = __builtin_amdgcn_wmma_f32_16x16x4_f32(false, a, false, b3, (short)0, acc3, false, false);
    }

    // D layout: reg r -> M = r + 8*half, N = l16
    #pragma unroll
    for (int j = 0; j < 4; ++j) {
        v8f acc = (j == 0) ? acc0 : (j == 1) ? acc1 : (j == 2) ? acc2 : acc3;
        int col = ch * 64 + j * 16 + l16;
        float bj = bias[col];
        #pragma unroll
        for (int r = 0; r < 8; ++r) {
            int row = mbase + rt * 16 + r + 8 * half;
            if (row < n) out[(size_t)row * D + col] = acc[r] + bj;
        }
    }
}

extern "C" void kernel_launch(void* const* d_in, const int* in_sizes, int n_in,
                              void* d_out, int out_size, void* d_ws, size_t ws_size,
                              hipStream_t stream) {
    // inputs: V (unused), E (2 x N_EDGES int), X (N x 128 f32), W (128 x 128 f32), b (128 f32)
    const int*   E    = (const int*)d_in[1];
    const int*   src  = E;               // E[0]
    const int*   dst  = E + N_EDGES;     // E[1]
    const float* X    = (const float*)d_in[2];
    const float* W    = (const float*)d_in[3];
    const float* bias = (const float*)d_in[4];

    float* Ha  = (float*)d_ws;                       // N*D floats ping buffer
    float* deg = Ha + (size_t)N_NODES * D;           // N floats: deg, then dinv in-place
    float* Hb  = (float*)d_out;                      // pong buffer, finally the output

    const int n = N_NODES, ne = N_EDGES;

    k_deg_init<<<(n + 255) / 256, 256, 0, stream>>>(deg, n);
    k_deg_count<<<(ne + 255) / 256, 256, 0, stream>>>(dst, deg, ne);
    k_dinv<<<(n + 255) / 256, 256, 0, stream>>>(deg, n);

    dim3 selfGrid((n * 32 + 255) / 256);             // 1 float4 per thread
    dim3 edgeGrid((ne * 32 + 255) / 256);            // 1 edge per wave32

    // hop 1: X -> Ha
    k_self_init<<<selfGrid, 256, 0, stream>>>(X, deg, Ha, n);
    k_edge_scatter<<<edgeGrid, 256, 0, stream>>>(src, dst, deg, X, Ha, ne);
    // hop 2: Ha -> Hb
    k_self_init<<<selfGrid, 256, 0, stream>>>(Ha, deg, Hb, n);
    k_edge_scatter<<<edgeGrid, 256, 0, stream>>>(src, dst, deg, Ha, Hb, ne);
    // hop 3: Hb -> Ha
    k_self_init<<<selfGrid, 256, 0, stream>>>(Hb, deg, Ha, n);
    k_edge_scatter<<<edgeGrid, 256, 0, stream>>>(src, dst, deg, Hb, Ha, ne);

    // Out = Ha @ W^T + b   (f32 WMMA)
    k_gemm_wmma<<<(n + 63) / 64, 256, 0, stream>>>(Ha, W, bias, (float*)d_out, n);
}